// GAT_75539884802468
// MI455X (gfx1250) — compile-verified
//
#include <hip/hip_runtime.h>
#include <hip/hip_bf16.h>

typedef __attribute__((ext_vector_type(2))) float v2f;
typedef __attribute__((ext_vector_type(8))) float v8f;

// -------- float -> order-preserving uint key (for atomicMax-based segment max)
__device__ __forceinline__ unsigned fkey(float f) {
    unsigned u = __float_as_uint(f);
    return (u & 0x80000000u) ? ~u : (u | 0x80000000u);
}
__device__ __forceinline__ float funkey(unsigned k) {
    unsigned u = (k & 0x80000000u) ? (k & 0x7fffffffu) : ~k;
    return __uint_as_float(u);
}

// ============================================================================
// GEMM: Out[M,256] = A[M,256] @ W[256,256]^T   (Out[m][o] = sum_k A[m][k]*W[o][k])
// Block = 128 threads = 4 waves. Block owns 16 rows; wave w owns cols [64w,64w+64).
// Uses V_WMMA_F32_16X16X4_F32, fp32 end-to-end.
// A-frag (16x4 f32): lane L holds M=L&15, K = 2*(L>>4) + v  -> contiguous b64 load.
// B-frag (4x16 f32): lane L holds N=L&15, K = 2*(L>>4) + v  -> W[o][k] row, b64 load.
// C/D (16x16 f32): vgpr j holds M = j + 8*(L>>4), N = L&15.
// ============================================================================
template <bool RELU_A>
__global__ __launch_bounds__(128) void gemm_nt_wmma(const float* __restrict__ A,
                                                    const float* __restrict__ W,
                                                    float* __restrict__ Out, int M) {
    const int K = 256;
    const int lane = threadIdx.x & 31;
    const int wave = threadIdx.x >> 5;
    const int n0 = wave * 64;
    const int m_base = blockIdx.x * 16;
    int mrow = m_base + (lane & 15);
    if (mrow >= M) mrow = M - 1;  // defensive (M % 16 == 0 in practice)
    const int khalf = (lane >> 4) * 2;
    const int col = lane & 15;

    v8f c0 = {}, c1 = {}, c2 = {}, c3 = {};
    const float* arow = A + (size_t)mrow * K;
    const float* w0 = W + (size_t)(n0 + 0 * 16 + col) * K;
    const float* w1 = W + (size_t)(n0 + 1 * 16 + col) * K;
    const float* w2 = W + (size_t)(n0 + 2 * 16 + col) * K;
    const float* w3 = W + (size_t)(n0 + 3 * 16 + col) * K;

#pragma unroll 4
    for (int k0 = 0; k0 < K; k0 += 4) {
        v2f a = *(const v2f*)(arow + k0 + khalf);
        if (RELU_A) {
            a.x = a.x > 0.f ? a.x : 0.f;
            a.y = a.y > 0.f ? a.y : 0.f;
        }
        v2f b0 = *(const v2f*)(w0 + k0 + khalf);
        v2f b1 = *(const v2f*)(w1 + k0 + khalf);
        v2f b2 = *(const v2f*)(w2 + k0 + khalf);
        v2f b3 = *(const v2f*)(w3 + k0 + khalf);
        c0 = __builtin_amdgcn_wmma_f32_16x16x4_f32(false, a, false, b0, (short)0, c0, false, false);
        c1 = __builtin_amdgcn_wmma_f32_16x16x4_f32(false, a, false, b1, (short)0, c1, false, false);
        c2 = __builtin_amdgcn_wmma_f32_16x16x4_f32(false, a, false, b2, (short)0, c2, false, false);
        c3 = __builtin_amdgcn_wmma_f32_16x16x4_f32(false, a, false, b3, (short)0, c3, false, false);
    }

    const int rbase = m_base + 8 * (lane >> 4);
#pragma unroll
    for (int j = 0; j < 8; ++j) {
        int r = rbase + j;
        if (r < M) {
            float* orow = Out + (size_t)r * 256 + n0 + col;
            orow[0]  = c0[j];
            orow[16] = c1[j];
            orow[32] = c2[j];
            orow[48] = c3[j];
        }
    }
}

// ============================================================================
// Per-node attention dot products: s_src[n,h] = dot(hp[n, h*FH:(h+1)*FH], a_src[h])
// One wave per node; lane covers 8 consecutive features; shfl_xor reduce per head.
// a_src/a_dst are flat length NH*FH == 256, matching feature index directly.
// ============================================================================
template <int NH, int FH>
__global__ void node_scores(const float* __restrict__ hp, const float* __restrict__ a_src,
                            const float* __restrict__ a_dst, float* __restrict__ s_src,
                            float* __restrict__ s_dst, int Nn) {
    int wid = (blockIdx.x * blockDim.x + threadIdx.x) >> 5;
    int lane = threadIdx.x & 31;
    if (wid >= Nn) return;
    const float* row = hp + (size_t)wid * 256;
    const int f0 = lane * 8;
    float ps = 0.f, pd = 0.f;
#pragma unroll
    for (int i = 0; i < 8; ++i) {
        float v = row[f0 + i];
        ps += v * a_src[f0 + i];
        pd += v * a_dst[f0 + i];
    }
    const int GL = FH / 8;  // lanes per head group (8 or 32)
#pragma unroll
    for (int off = 1; off < GL; off <<= 1) {
        ps += __shfl_xor(ps, off, 32);
        pd += __shfl_xor(pd, off, 32);
    }
    if ((lane & (GL - 1)) == 0) {
        int h = f0 / FH;
        s_src[(size_t)wid * NH + h] = ps;
        s_dst[(size_t)wid * NH + h] = pd;
    }
}

// ============================================================================
// Edge pass 1: leaky-relu score, per-(dst,head) running max via uint-key atomicMax
// ============================================================================
template <int NH>
__global__ void edge_max_kernel(const int* __restrict__ src, const int* __restrict__ dst,
                                const float* __restrict__ ssrc, const float* __restrict__ sdst,
                                unsigned* __restrict__ emax, int E) {
    int i = blockIdx.x * blockDim.x + threadIdx.x;
    if (i >= E) return;
    int s = src[i], d = dst[i];
#pragma unroll
    for (int h = 0; h < NH; ++h) {
        float e = ssrc[(size_t)s * NH + h] + sdst[(size_t)d * NH + h];
        e = e > 0.f ? e : 0.2f * e;
        atomicMax(&emax[(size_t)d * NH + h], fkey(e));
    }
}

// ============================================================================
// Edge pass 2: ex = exp(e - max[dst]); store per-edge, atomicAdd per-(dst,head) sum
// ============================================================================
template <int NH>
__global__ void edge_expsum_kernel(const int* __restrict__ src, const int* __restrict__ dst,
                                   const float* __restrict__ ssrc, const float* __restrict__ sdst,
                                   const unsigned* __restrict__ emax, float* __restrict__ exb,
                                   float* __restrict__ esum, int E) {
    int i = blockIdx.x * blockDim.x + threadIdx.x;
    if (i >= E) return;
    int s = src[i], d = dst[i];
#pragma unroll
    for (int h = 0; h < NH; ++h) {
        float e = ssrc[(size_t)s * NH + h] + sdst[(size_t)d * NH + h];
        e = e > 0.f ? e : 0.2f * e;
        float m = funkey(emax[(size_t)d * NH + h]);
        float ex = __expf(e - m);
        exb[(size_t)i * NH + h] = ex;
        atomicAdd(&esum[(size_t)d * NH + h], ex);
    }
}

// ============================================================================
// Edge pass 3: out[dst, :] += hp[src, :] * alpha[head].  One wave per edge,
// 8 floats per lane (one head per lane for NH=4): coalesced b128 gather + f32 atomics.
// ============================================================================
template <int NH, int FH>
__global__ void edge_aggregate(const int* __restrict__ src, const int* __restrict__ dst,
                               const float* __restrict__ hp, const float* __restrict__ exb,
                               const float* __restrict__ esum, float* __restrict__ out, int E) {
    int wid = (blockIdx.x * blockDim.x + threadIdx.x) >> 5;
    int lane = threadIdx.x & 31;
    if (wid >= E) return;
    int s = src[wid], d = dst[wid];
    const int f0 = lane * 8;
    const int h = f0 / FH;  // constant across a lane's 8 features (8 | FH)
    float alpha = exb[(size_t)wid * NH + h] / (esum[(size_t)d * NH + h] + 1e-16f);
    const float* hrow = hp + (size_t)s * 256 + f0;
    float* orow = out + (size_t)d * 256 + f0;
    float4 v0 = *(const float4*)(hrow);
    float4 v1 = *(const float4*)(hrow + 4);
    atomicAdd(orow + 0, v0.x * alpha);
    atomicAdd(orow + 1, v0.y * alpha);
    atomicAdd(orow + 2, v0.z * alpha);
    atomicAdd(orow + 3, v0.w * alpha);
    atomicAdd(orow + 4, v1.x * alpha);
    atomicAdd(orow + 5, v1.y * alpha);
    atomicAdd(orow + 6, v1.z * alpha);
    atomicAdd(orow + 7, v1.w * alpha);
}

// ============================================================================
extern "C" void kernel_launch(void* const* d_in, const int* in_sizes, int n_in,
                              void* d_out, int out_size, void* d_ws, size_t ws_size,
                              hipStream_t stream) {
    (void)n_in; (void)out_size; (void)ws_size;
    const float* x   = (const float*)d_in[0];
    const int*   ei  = (const int*)d_in[1];
    const float* W1  = (const float*)d_in[2];
    const float* a1s = (const float*)d_in[3];
    const float* a1d = (const float*)d_in[4];
    const float* W2  = (const float*)d_in[5];
    const float* a2s = (const float*)d_in[6];
    const float* a2d = (const float*)d_in[7];

    const int Nn = in_sizes[0] / 256;
    const int E  = in_sizes[1] / 2;
    const int* src = ei;
    const int* dst = ei + E;

    // --- workspace carve-out (hp reused across layers; ~118 MB total) ---
    char* ws = (char*)d_ws;
    size_t off = 0;
    auto alloc = [&](size_t bytes) -> void* {
        void* p = ws + off;
        off += (bytes + 255) & ~(size_t)255;
        return p;
    };
    float*    hp   = (float*)alloc((size_t)Nn * 256 * sizeof(float));   // hp1, then hp2
    float*    h1   = (float*)alloc((size_t)Nn * 256 * sizeof(float));   // layer-1 aggregate
    float*    ssrc = (float*)alloc((size_t)Nn * 4 * sizeof(float));
    float*    sdst = (float*)alloc((size_t)Nn * 4 * sizeof(float));
    unsigned* emax = (unsigned*)alloc((size_t)Nn * 4 * sizeof(unsigned));
    float*    esum = (float*)alloc((size_t)Nn * 4 * sizeof(float));
    float*    exb  = (float*)alloc((size_t)E * 4 * sizeof(float));
    float*    outF = (float*)d_out;

    const int mtiles      = (Nn + 15) / 16;
    const int node_blocks = (Nn + 7) / 8;                 // wave per node, 8 waves/block
    const int edge_blocks = (E + 255) / 256;              // thread per edge
    const int eagg_blocks = (int)(((size_t)E * 32 + 255) / 256);  // wave per edge

    // =================== Layer 1 (4 heads x 64) ===================
    gemm_nt_wmma<false><<<mtiles, 128, 0, stream>>>(x, W1, hp, Nn);
    node_scores<4, 64><<<node_blocks, 256, 0, stream>>>(hp, a1s, a1d, ssrc, sdst, Nn);
    hipMemsetAsync(emax, 0, (size_t)Nn * 4 * sizeof(unsigned), stream);
    hipMemsetAsync(esum, 0, (size_t)Nn * 4 * sizeof(float), stream);
    hipMemsetAsync(h1,   0, (size_t)Nn * 256 * sizeof(float), stream);
    edge_max_kernel<4><<<edge_blocks, 256, 0, stream>>>(src, dst, ssrc, sdst, emax, E);
    edge_expsum_kernel<4><<<edge_blocks, 256, 0, stream>>>(src, dst, ssrc, sdst, emax, exb, esum, E);
    edge_aggregate<4, 64><<<eagg_blocks, 256, 0, stream>>>(src, dst, hp, exb, esum, h1, E);

    // =================== Layer 2 (1 head x 256), ReLU fused into A-load ===================
    gemm_nt_wmma<true><<<mtiles, 128, 0, stream>>>(h1, W2, hp, Nn);
    node_scores<1, 256><<<node_blocks, 256, 0, stream>>>(hp, a2s, a2d, ssrc, sdst, Nn);
    hipMemsetAsync(emax, 0, (size_t)Nn * sizeof(unsigned), stream);
    hipMemsetAsync(esum, 0, (size_t)Nn * sizeof(float), stream);
    hipMemsetAsync(d_out, 0, (size_t)Nn * 256 * sizeof(float), stream);
    edge_max_kernel<1><<<edge_blocks, 256, 0, stream>>>(src, dst, ssrc, sdst, emax, E);
    edge_expsum_kernel<1><<<edge_blocks, 256, 0, stream>>>(src, dst, ssrc, sdst, emax, exb, esum, E);
    edge_aggregate<1, 256><<<eagg_blocks, 256, 0, stream>>>(src, dst, hp, exb, esum, outF, E);
}